// GraphNeuralNetwork_22677427323618
// MI455X (gfx1250) — compile-verified
//
#include <hip/hip_runtime.h>
#include <math.h>

typedef float v2f __attribute__((ext_vector_type(2)));
typedef float v4f __attribute__((ext_vector_type(4)));
typedef float v8f __attribute__((ext_vector_type(8)));

#define GCN_NODES 100000
#define GCN_H 256

// ---------------------------------------------------------------------------
// fp32 GEMM via V_WMMA_F32_16X16X4_F32.
// C[M,N] = A[M,K] @ B[K,N] (+ bias). Row-major. N % 32 == 0, K % 16 == 0,
// K even. Block: 256 threads = 8 waves. Wave tile 64(M) x 32(N) -> 8
// accumulators, 8 wmma per K-step of 4 against 8 vmem loads.
//   lognwn=3: block tile  64M x 256N (all 8 waves share one A row set)
//   lognwn=2: block tile 128M x 128N
// All loads: uniform SGPR base + 32-bit unsigned element offset (saddr +
// scale_offset form); A is indexed in v2f units to keep b64 loads on that path.
// ---------------------------------------------------------------------------
__global__ __launch_bounds__(256) void gemm_f32_wmma(
    const float* __restrict__ A, const float* __restrict__ B,
    const float* __restrict__ bias, float* __restrict__ C,
    int M, int K, int N, int lognwn)
{
    const int tid  = threadIdx.x;
    const int lane = tid & 31;
    const int wave = tid >> 5;
    const int nwn  = 1 << lognwn;              // N-waves per block
    const int wm   = wave >> lognwn;           // M-wave index
    const int wn   = wave & (nwn - 1);         // N-wave index
    const int waveM = (blockIdx.y * (8 >> lognwn) + wm) * 64;
    const int waveN = blockIdx.x * (nwn * 32) + wn * 32;

    const int lm    = lane & 15;
    const int khalf = lane >> 4;               // 0 or 1

    // A fragment offsets (v2f units) for 4 M-subtiles (rows clamped: every
    // lane issues a valid load, OOB rows discarded at store -> EXEC all-1s).
    const v2f* A2 = (const v2f*)A;
    unsigned aoff2[4];
#pragma unroll
    for (int mi = 0; mi < 4; ++mi) {
        int r = waveM + mi * 16 + lm;
        if (r >= M) r = M - 1;
        aoff2[mi] = ((unsigned)r * (unsigned)K + 2u * (unsigned)khalf) >> 1;
    }

    // B fragment column offsets (float units) for 2 N-subtiles.
    const unsigned boff0 = (unsigned)(2 * khalf) * (unsigned)N
                         + (unsigned)(waveN + lm);
    const unsigned boff1 = boff0 + 16u;

    v8f acc[4][2];
#pragma unroll
    for (int mi = 0; mi < 4; ++mi) { acc[mi][0] = (v8f){}; acc[mi][1] = (v8f){}; }

#pragma unroll 4
    for (int k = 0; k < K; k += 4) {
        const unsigned kv = (unsigned)k >> 1;          // v2f units
        // A 16x4 frags: lane holds A[m][k+2*khalf], A[m][k+2*khalf+1]
        v2f a[4];
#pragma unroll
        for (int mi = 0; mi < 4; ++mi)
            a[mi] = A2[(size_t)(aoff2[mi] + kv)];
        // B 4x16 frags: lane holds B[k+2*khalf][n], B[k+2*khalf+1][n]
        const unsigned bk = (unsigned)k * (unsigned)N;
        v2f b0, b1;
        b0.x = B[(size_t)(boff0 + bk)];
        b0.y = B[(size_t)(boff0 + bk + (unsigned)N)];
        b1.x = B[(size_t)(boff1 + bk)];
        b1.y = B[(size_t)(boff1 + bk + (unsigned)N)];

#pragma unroll
        for (int mi = 0; mi < 4; ++mi) {
            acc[mi][0] = __builtin_amdgcn_wmma_f32_16x16x4_f32(
                false, a[mi], false, b0, (short)0, acc[mi][0], false, false);
            acc[mi][1] = __builtin_amdgcn_wmma_f32_16x16x4_f32(
                false, a[mi], false, b1, (short)0, acc[mi][1], false, false);
        }
    }

    // C/D layout: lane col = lane&15; VGPR r = row r (lanes 0-15) / r+8 (16-31)
    const unsigned c0 = (unsigned)(waveN + lm);
    const unsigned c1 = c0 + 16u;
    float bv0 = 0.f, bv1 = 0.f;
    if (bias) { bv0 = bias[c0]; bv1 = bias[c1]; }

#pragma unroll
    for (int mi = 0; mi < 4; ++mi) {
#pragma unroll
        for (int r = 0; r < 8; ++r) {
            int row = waveM + mi * 16 + r + 8 * khalf;
            if (row < M) {
                unsigned o = (unsigned)row * (unsigned)N;
                C[(size_t)(o + c0)] = acc[mi][0][r] + bv0;
                C[(size_t)(o + c1)] = acc[mi][1][r] + bv1;
            }
        }
    }
}

// ---------------------------------------------------------------------------
// Degree / norm helpers. deg starts at 1.0 (self loop), +1 per incoming edge,
// then converted in-place to deg^-1/2.
// ---------------------------------------------------------------------------
__global__ __launch_bounds__(256) void deg_init(float* __restrict__ deg, int n)
{
    int i = blockIdx.x * 256 + threadIdx.x;
    if (i < n) deg[i] = 1.0f;
}

__global__ __launch_bounds__(256) void deg_count(
    const long long* __restrict__ dst, float* __restrict__ deg, int E)
{
    int e = blockIdx.x * 256 + threadIdx.x;
    if (e < E) atomicAdd(&deg[(unsigned)dst[e]], 1.0f);
}

__global__ __launch_bounds__(256) void deg_rsqrt(float* __restrict__ deg, int n)
{
    int i = blockIdx.x * 256 + threadIdx.x;
    if (i < n) deg[i] = rsqrtf(deg[i]);   // deg >= 1 always (self loop)
}

// ---------------------------------------------------------------------------
// Edge aggregation: acc[dst] += xw[src] * dis[src]*dis[dst].
// 64 threads per edge, 4 edges per 256-thread block; one float4 per thread,
// scatter with global f32 atomics (102MB accumulator is L2-resident: <192MB).
// ---------------------------------------------------------------------------
__global__ __launch_bounds__(256) void aggregate_edges(
    const long long* __restrict__ src, const long long* __restrict__ dst,
    const float* __restrict__ dis, const float* __restrict__ xw,
    float* __restrict__ acc, int E)
{
    int e = blockIdx.x * 4 + (threadIdx.x >> 6);
    if (e >= E) return;
    unsigned lane = threadIdx.x & 63;          // covers 64 float4 = 256 cols
    unsigned s = (unsigned)src[e];
    unsigned d = (unsigned)dst[e];
    float nrm = dis[s] * dis[d];
    v4f v = *(const v4f*)(xw + (size_t)(s * (unsigned)GCN_H + lane * 4u));
    float* o = acc + (size_t)(d * (unsigned)GCN_H + lane * 4u);
    atomicAdd(o + 0, v.x * nrm);
    atomicAdd(o + 1, v.y * nrm);
    atomicAdd(o + 2, v.z * nrm);
    atomicAdd(o + 3, v.w * nrm);
}

// ---------------------------------------------------------------------------
// Epilogue: fold in the self-loop term (norm = dis[i]^2), bias, ReLU.
// acc[i] = relu(acc[i] + xw[i]*dis[row]^2 + b[col]), in place.
// ---------------------------------------------------------------------------
__global__ __launch_bounds__(256) void epilogue_selfloop_bias_relu(
    float* __restrict__ acc, const float* __restrict__ xw,
    const float* __restrict__ dis, const float* __restrict__ b, int n)
{
    int i = blockIdx.x * 256 + threadIdx.x;
    if (i >= n) return;
    unsigned u = (unsigned)i;
    unsigned row = u >> 8;          // GCN_H == 256
    unsigned col = u & 255u;
    float di = dis[row];
    float v = acc[u] + xw[u] * (di * di) + b[col];
    acc[u] = v > 0.f ? v : 0.f;
}

// ---------------------------------------------------------------------------
extern "C" void kernel_launch(void* const* d_in, const int* in_sizes, int n_in,
                              void* d_out, int out_size, void* d_ws, size_t ws_size,
                              hipStream_t stream)
{
    const float*     x   = (const float*)d_in[0];
    const long long* ei  = (const long long*)d_in[1];   // int64 [2, E]
    const float*     W1  = (const float*)d_in[2];
    const float*     b1  = (const float*)d_in[3];
    const float*     W2  = (const float*)d_in[4];
    const float*     b2  = (const float*)d_in[5];
    const float*     fcw = (const float*)d_in[6];
    const float*     fcb = (const float*)d_in[7];
    float*           out = (float*)d_out;

    const int E = in_sizes[1] / 2;
    const long long* src = ei;          // edge_index[0]
    const long long* dst = ei + E;      // edge_index[1]

    const size_t NH = (size_t)GCN_NODES * GCN_H;        // 25.6M floats
    char*  ws   = (char*)d_ws;
    float* buf0 = (float*)ws;                           // xw (GEMM output)
    float* buf1 = (float*)(ws + NH * sizeof(float));    // aggregation acc / h
    float* dis  = (float*)(ws + 2 * NH * sizeof(float)); // deg -> deg^-1/2

    const dim3 blk(256);

    // ---- normalization coefficients -------------------------------------
    deg_init <<<(GCN_NODES + 255) / 256, blk, 0, stream>>>(dis, GCN_NODES);
    deg_count<<<(E + 255) / 256,        blk, 0, stream>>>(dst, dis, E);
    deg_rsqrt<<<(GCN_NODES + 255) / 256, blk, 0, stream>>>(dis, GCN_NODES);

    // ---- layer 1: buf0 = x @ W1 ; aggregate ; relu(+b1) -> buf1 ----------
    {   // N=256 -> 8 N-waves (lognwn=3), block tile 64x256
        dim3 grid(1, (GCN_NODES + 63) / 64);
        gemm_f32_wmma<<<grid, blk, 0, stream>>>(x, W1, nullptr, buf0,
                                                GCN_NODES, 512, 256, 3);
    }
    hipMemsetAsync(buf1, 0, NH * sizeof(float), stream);
    aggregate_edges<<<(E + 3) / 4, blk, 0, stream>>>(src, dst, dis, buf0, buf1, E);
    epilogue_selfloop_bias_relu<<<(int)((NH + 255) / 256), blk, 0, stream>>>(
        buf1, buf0, dis, b1, (int)NH);

    // ---- layer 2: buf0 = buf1 @ W2 ; aggregate ; relu(+b2) -> buf1 -------
    {
        dim3 grid(1, (GCN_NODES + 63) / 64);
        gemm_f32_wmma<<<grid, blk, 0, stream>>>(buf1, W2, nullptr, buf0,
                                                GCN_NODES, 256, 256, 3);
    }
    hipMemsetAsync(buf1, 0, NH * sizeof(float), stream);
    aggregate_edges<<<(E + 3) / 4, blk, 0, stream>>>(src, dst, dis, buf0, buf1, E);
    epilogue_selfloop_bias_relu<<<(int)((NH + 255) / 256), blk, 0, stream>>>(
        buf1, buf0, dis, b2, (int)NH);

    // ---- FC head: out = buf1 @ fcw + fcb ---------------------------------
    {   // N=128 -> 4 N-waves (lognwn=2), block tile 128x128
        dim3 grid(1, (GCN_NODES + 127) / 128);
        gemm_f32_wmma<<<grid, blk, 0, stream>>>(buf1, fcw, fcb, out,
                                                GCN_NODES, 256, 128, 2);
    }
}